// GCN_2740189135451
// MI455X (gfx1250) — compile-verified
//
#include <hip/hip_runtime.h>
#include <hip/hip_bf16.h>

// GCN 3-layer forward for MI455X (gfx1250, wave32).
// GEMMs use V_WMMA_F32_16X16X4_F32 (exact fp32, GEMM is ~4 GFLOP total - not
// the bottleneck). Edge aggregation is L2-resident gather + hardware fp32
// atomics (global_atomic_add_f32), which dominates: ~850K edges x 128 feats.

#define GCN_N 50000

typedef __attribute__((ext_vector_type(2))) float v2f;
typedef __attribute__((ext_vector_type(8))) float v8f;

// ---------------- degree / normalization ----------------

__global__ void deg_init_kernel(float* __restrict__ deg, int n) {
  int i = blockIdx.x * blockDim.x + threadIdx.x;
  if (i < n) deg[i] = 1.0f;  // self-loop contribution
}

__global__ void deg_count_kernel(const int* __restrict__ dst,
                                 float* __restrict__ deg, int numE) {
  int e = blockIdx.x * blockDim.x + threadIdx.x;
  if (e < numE) unsafeAtomicAdd(&deg[dst[e]], 1.0f);
}

__global__ void dinv_kernel(const float* __restrict__ deg,
                            float* __restrict__ dinv, int n) {
  int i = blockIdx.x * blockDim.x + threadIdx.x;
  if (i < n) dinv[i] = rsqrtf(deg[i]);  // deg >= 1 always
}

// ---------------- WMMA fp32 GEMM: T[n x DOUT] = H[n x 128] @ W[128 x DOUT] --

// Block = 256 threads = 8 waves; each wave owns one 16x16 output tile.
// A layout (16x4 f32): lane L: m = L%16, half = L/16; vgpr v holds K = k0 + 2*half + v.
// B layout (4x16 f32): lane L: n = L%16, half = L/16; vgpr v holds K = k0 + 2*half + v.
// D layout (16x16 f32): vgpr r holds row r (lanes 0-15) / row r+8 (lanes 16-31), col = L%16.
template <int DOUT>
__global__ void __launch_bounds__(256)
gemm128_wmma_kernel(const float* __restrict__ H, const float* __restrict__ W,
                    float* __restrict__ T, int nrows) {
  __shared__ float Ws[128 * 16];  // 16-wide column strip of W, row-major
  const int tid  = threadIdx.x;
  const int lane = tid & 31;
  const int wave = tid >> 5;
  const int n0   = blockIdx.y * 16;

  for (int i = tid; i < 128 * 16; i += 256) {
    int k = i >> 4, nn = i & 15;
    Ws[i] = W[k * DOUT + n0 + nn];
  }
  __syncthreads();

  const int m0 = (blockIdx.x * 8 + wave) * 16;
  if (m0 >= nrows) return;  // wave-uniform: EXEC stays all-ones for WMMA

  const int ln   = lane & 15;  // m for A, n for B/D
  const int half = lane >> 4;
  const float* arow = H + (size_t)(m0 + ln) * 128;

  v8f acc = {0.f, 0.f, 0.f, 0.f, 0.f, 0.f, 0.f, 0.f};
#pragma unroll
  for (int k0 = 0; k0 < 128; k0 += 4) {
    v2f a = *(const v2f*)(arow + k0 + 2 * half);  // global_load_b64, 8B aligned
    v2f b;
    b.x = Ws[(k0 + 2 * half + 0) * 16 + ln];      // conflict-free ds_load_b32
    b.y = Ws[(k0 + 2 * half + 1) * 16 + ln];
    acc = __builtin_amdgcn_wmma_f32_16x16x4_f32(
        /*neg_a=*/false, a, /*neg_b=*/false, b,
        /*c_mod=*/(short)0, acc, /*reuse_a=*/false, /*reuse_b=*/false);
  }

  float* dtile = T + (size_t)m0 * DOUT + n0;
#pragma unroll
  for (int r = 0; r < 8; ++r)
    dtile[(size_t)(r + 8 * half) * DOUT + ln] = acc[r];
}

// ---------------- aggregation ----------------

// agg[i][:] = T[i][:] * dinv[i]^2   (self-loop term; also zero-initializes agg)
template <int LGD>
__global__ void self_init_kernel(const float* __restrict__ T,
                                 const float* __restrict__ dinv,
                                 float* __restrict__ agg, int n) {
  int i = blockIdx.x * blockDim.x + threadIdx.x;
  int total = n << LGD;
  if (i >= total) return;
  int row = i >> LGD;
  float w = dinv[row] * dinv[row];
  agg[i] = T[i] * w;
}

// agg[dst][:] += T[src][:] * dinv[src]*dinv[dst]  via hw fp32 atomics.
// One thread per (edge, float4 chunk); a full wave covers one edge when D=128.
template <int LGC>  // chunks per edge = 1<<LGC, D = 4<<LGC
__global__ void edge_agg_kernel(const float* __restrict__ T,
                                const int* __restrict__ srcIdx,
                                const int* __restrict__ dstIdx,
                                const float* __restrict__ dinv,
                                float* __restrict__ agg, int numE) {
  int i = blockIdx.x * blockDim.x + threadIdx.x;
  int total = numE << LGC;
  if (i >= total) return;
  int e = i >> LGC;
  int c = i & ((1 << LGC) - 1);
  int s = srcIdx[e], d = dstIdx[e];
  float w = dinv[s] * dinv[d];
  const int D = 4 << LGC;
  const float4 v = *(const float4*)(T + (size_t)s * D + (c << 2));
  float* o = agg + (size_t)d * D + (c << 2);
  unsafeAtomicAdd(o + 0, v.x * w);
  unsafeAtomicAdd(o + 1, v.y * w);
  unsafeAtomicAdd(o + 2, v.z * w);
  unsafeAtomicAdd(o + 3, v.w * w);
}

// out[i][j] = act(agg[i][j] + bias[j])
template <int LGD, bool RELU>
__global__ void bias_act_kernel(const float* __restrict__ agg,
                                const float* __restrict__ bias,
                                float* __restrict__ out, int n) {
  int i = blockIdx.x * blockDim.x + threadIdx.x;
  int total = n << LGD;
  if (i >= total) return;
  float v = agg[i] + bias[i & ((1 << LGD) - 1)];
  out[i] = RELU ? fmaxf(v, 0.0f) : v;
}

// ---------------- launch ----------------

extern "C" void kernel_launch(void* const* d_in, const int* in_sizes, int n_in,
                              void* d_out, int out_size, void* d_ws, size_t ws_size,
                              hipStream_t stream) {
  const float* x  = (const float*)d_in[0];
  const int*   ei = (const int*)d_in[1];
  const float* W1 = (const float*)d_in[2];
  const float* b1 = (const float*)d_in[3];
  const float* W2 = (const float*)d_in[4];
  const float* b2 = (const float*)d_in[5];
  const float* W3 = (const float*)d_in[6];
  const float* b3 = (const float*)d_in[7];

  const int Nn = GCN_N;
  const int Ee = in_sizes[1] / 2;
  const int* src = ei;
  const int* dst = ei + Ee;

  // Workspace layout (floats): deg[N] | dinv[N] | T[N*128] | A[N*128] | B[N*128]
  float* deg  = (float*)d_ws;
  float* dinv = deg + Nn;
  float* T    = dinv + Nn;
  float* A    = T + (size_t)Nn * 128;
  float* B    = A + (size_t)Nn * 128;

  const int TPB = 256;
  auto blk = [](int total) { return (total + 255) / 256; };

  // Normalization: deg = 1 + in-degree; dinv = rsqrt(deg)
  deg_init_kernel<<<blk(Nn), TPB, 0, stream>>>(deg, Nn);
  deg_count_kernel<<<blk(Ee), TPB, 0, stream>>>(dst, deg, Ee);
  dinv_kernel<<<blk(Nn), TPB, 0, stream>>>(deg, dinv, Nn);

  const dim3 g128((Nn / 16 + 7) / 8, 128 / 16);  // 391 x 8
  const dim3 g64((Nn / 16 + 7) / 8, 64 / 16);    // 391 x 4

  // ---- Layer 1: x -> A (relu) ----
  gemm128_wmma_kernel<128><<<g128, TPB, 0, stream>>>(x, W1, T, Nn);
  self_init_kernel<7><<<blk(Nn << 7), TPB, 0, stream>>>(T, dinv, A, Nn);
  edge_agg_kernel<5><<<blk(Ee << 5), TPB, 0, stream>>>(T, src, dst, dinv, A, Ee);
  bias_act_kernel<7, true><<<blk(Nn << 7), TPB, 0, stream>>>(A, b1, A, Nn);

  // ---- Layer 2: A -> B (relu) ----
  gemm128_wmma_kernel<128><<<g128, TPB, 0, stream>>>(A, W2, T, Nn);
  self_init_kernel<7><<<blk(Nn << 7), TPB, 0, stream>>>(T, dinv, B, Nn);
  edge_agg_kernel<5><<<blk(Ee << 5), TPB, 0, stream>>>(T, src, dst, dinv, B, Ee);
  bias_act_kernel<7, true><<<blk(Nn << 7), TPB, 0, stream>>>(B, b2, B, Nn);

  // ---- Layer 3: B -> d_out (no relu) ----
  gemm128_wmma_kernel<64><<<g64, TPB, 0, stream>>>(B, W3, T, Nn);
  self_init_kernel<6><<<blk(Nn << 6), TPB, 0, stream>>>(T, dinv, A, Nn);
  edge_agg_kernel<4><<<blk(Ee << 4), TPB, 0, stream>>>(T, src, dst, dinv, A, Ee);
  bias_act_kernel<6, false><<<blk(Nn << 6), TPB, 0, stream>>>(A, b3, (float*)d_out, Nn);
}